// GATEncoder_36575941492955
// MI455X (gfx1250) — compile-verified
//
#include <hip/hip_runtime.h>
#include <hip/hip_bf16.h>
#include <math.h>

// ---------------- problem constants (from reference) ----------------
#define NN       50000      // nodes
#define NE       800000     // edges (before self loops)
#define ETOT     850000     // edges + self loops
#define IN_CH    64
#define HIDC     32         // hidden channels per head
#define HEADS    8
#define H1C      256        // HEADS*HIDC
#define OUTC     64
#define NEG_SLOPE 0.2f

typedef __attribute__((ext_vector_type(2))) float v2f;
typedef __attribute__((ext_vector_type(8))) float v8f;

// ---------------- helpers ----------------
__device__ __forceinline__ unsigned f2ord(float f) {
    unsigned u = __float_as_uint(f);
    return (u & 0x80000000u) ? ~u : (u | 0x80000000u);
}
__device__ __forceinline__ float ord2f(unsigned u) {
    return (u & 0x80000000u) ? __uint_as_float(u & 0x7fffffffu)
                             : __uint_as_float(~u);
}
#define ORD_NEG_INF 0x007FFFFFu   // f2ord(-inf)

__device__ __forceinline__ float lrelu(float x) {
    return x > 0.0f ? x : NEG_SLOPE * x;
}

__device__ __forceinline__ void get_edge(const long long* __restrict__ ei, int e,
                                         int& s, int& d) {
    if (e < NE) { s = (int)ei[e]; d = (int)ei[NE + e]; }
    else        { s = e - NE; d = e - NE; }          // self loop
}

// ---------------- fills ----------------
__global__ void fill_f32(float* p, float v, int n) {
    int i = blockIdx.x * blockDim.x + threadIdx.x;
    if (i < n) p[i] = v;
}
__global__ void fill_u32(unsigned* p, unsigned v, int n) {
    int i = blockIdx.x * blockDim.x + threadIdx.x;
    if (i < n) p[i] = v;
}

// ================= GEMM1: C[M,256] = A[M,64] @ B[64,256] =================
// One wave per 16-row tile. A resident in regs (32 VGPRs), n-tile loop outer
// with a single v8f accumulator; all 16 B chunks for a tile batch-loaded so
// WMMAs issue back-to-back after one wait.
__global__ void gemm1_wmma(const float* __restrict__ A,
                           const float* __restrict__ B,
                           float* __restrict__ C, int M) {
    constexpr int K = IN_CH, NCOLS = H1C, CHUNKS = K / 4, NT = NCOLS / 16;
    const int wave = blockIdx.x * (blockDim.x >> 5) + (threadIdx.x >> 5);
    if (wave >= (M >> 4)) return;
    const int lane = threadIdx.x & 31;
    const int row  = lane & 15;
    const int khi  = lane >> 4;           // 0: k%4 in {0,1}; 1: k%4 in {2,3}
    const int m0   = wave << 4;

    // A: 16 chunks, each a contiguous pair -> single b64 loads
    v2f a[CHUNKS];
    const float* aRow = A + (size_t)(m0 + row) * K + 2 * khi;
    #pragma unroll
    for (int c = 0; c < CHUNKS; ++c)
        a[c] = *(const v2f*)(aRow + 4 * c);

    #pragma unroll
    for (int nt = 0; nt < NT; ++nt) {
        const float* bCol = B + nt * 16 + row;
        v2f b[CHUNKS];
        #pragma unroll
        for (int c = 0; c < CHUNKS; ++c) {
            const int k0 = 4 * c + 2 * khi;
            b[c].x = bCol[(size_t)k0 * NCOLS];
            b[c].y = bCol[(size_t)(k0 + 1) * NCOLS];
        }
        v8f acc = {};
        #pragma unroll
        for (int c = 0; c < CHUNKS; ++c)
            acc = __builtin_amdgcn_wmma_f32_16x16x4_f32(
                false, a[c], false, b[c], (short)0, acc, false, false);
        #pragma unroll
        for (int i = 0; i < 8; ++i)
            C[(size_t)(m0 + i + 8 * khi) * NCOLS + nt * 16 + row] = acc[i];
    }
}

// ================= GEMM2: C[M,64] = A[M,256] @ B[256,64] =================
// 4 resident accumulators (32 VGPRs); K processed in groups of 8 chunks with
// a[8] + b[8][4] batch-loaded (80 VGPRs) ahead of 32 WMMAs per group.
__global__ void gemm2_wmma(const float* __restrict__ A,
                           const float* __restrict__ B,
                           float* __restrict__ C, int M) {
    constexpr int K = H1C, NCOLS = OUTC, CHUNKS = K / 4, NT = NCOLS / 16;
    constexpr int CG = 8, NG = CHUNKS / CG;
    const int wave = blockIdx.x * (blockDim.x >> 5) + (threadIdx.x >> 5);
    if (wave >= (M >> 4)) return;
    const int lane = threadIdx.x & 31;
    const int row  = lane & 15;
    const int khi  = lane >> 4;
    const int m0   = wave << 4;

    v8f acc[NT] = {};
    const float* aRow = A + (size_t)(m0 + row) * K + 2 * khi;
    const float* bCol = B + row;

    for (int g = 0; g < NG; ++g) {
        v2f a[CG];
        v2f b[CG][NT];
        #pragma unroll
        for (int c = 0; c < CG; ++c) {
            const int cc = g * CG + c;
            a[c] = *(const v2f*)(aRow + 4 * cc);
            const int k0 = 4 * cc + 2 * khi;
            #pragma unroll
            for (int nt = 0; nt < NT; ++nt) {
                b[c][nt].x = bCol[(size_t)k0 * NCOLS + nt * 16];
                b[c][nt].y = bCol[(size_t)(k0 + 1) * NCOLS + nt * 16];
            }
        }
        #pragma unroll
        for (int c = 0; c < CG; ++c) {
            #pragma unroll
            for (int nt = 0; nt < NT; ++nt)
                acc[nt] = __builtin_amdgcn_wmma_f32_16x16x4_f32(
                    false, a[c], false, b[c][nt], (short)0, acc[nt], false, false);
        }
    }

    #pragma unroll
    for (int nt = 0; nt < NT; ++nt)
        #pragma unroll
        for (int i = 0; i < 8; ++i)
            C[(size_t)(m0 + i + 8 * khi) * NCOLS + nt * 16 + row] = acc[nt][i];
}

// ---------------- attention logits ----------------
__global__ void attn_logits1(const float* __restrict__ h1,
                             const float* __restrict__ a_s,
                             const float* __restrict__ a_d,
                             float* __restrict__ als, float* __restrict__ ald) {
    int tid = blockIdx.x * blockDim.x + threadIdx.x;
    if (tid >= NN * HEADS) return;
    int n = tid >> 3, h = tid & 7;
    const float* hp = h1 + (size_t)n * H1C + h * HIDC;
    const float* as = a_s + h * HIDC;
    const float* ad = a_d + h * HIDC;
    float ss = 0.f, sd = 0.f;
    #pragma unroll
    for (int c = 0; c < HIDC; ++c) { float v = hp[c]; ss += v * as[c]; sd += v * ad[c]; }
    als[tid] = ss;
    ald[tid] = sd;
}

__global__ void attn_logits2(const float* __restrict__ h2,
                             const float* __restrict__ a_s,
                             const float* __restrict__ a_d,
                             float* __restrict__ als, float* __restrict__ ald) {
    int n = blockIdx.x * blockDim.x + threadIdx.x;
    if (n >= NN) return;
    const float* hp = h2 + (size_t)n * OUTC;
    float ss = 0.f, sd = 0.f;
    #pragma unroll
    for (int c = 0; c < OUTC; ++c) { float v = hp[c]; ss += v * a_s[c]; sd += v * a_d[c]; }
    als[n] = ss;
    ald[n] = sd;
}

// ---------------- edge passes (softmax over incoming edges of dst) ----------------
template<int HEADS_T>
__global__ void edge_max(const long long* __restrict__ ei,
                         const float* __restrict__ als, const float* __restrict__ ald,
                         unsigned* __restrict__ mbuf) {
    int tid = blockIdx.x * blockDim.x + threadIdx.x;
    if (tid >= ETOT * HEADS_T) return;
    int e = tid / HEADS_T, h = tid % HEADS_T;
    int s, d; get_edge(ei, e, s, d);
    float ev = lrelu(als[s * HEADS_T + h] + ald[d * HEADS_T + h]);
    atomicMax(&mbuf[d * HEADS_T + h], f2ord(ev));
}

template<int HEADS_T>
__global__ void edge_denom(const long long* __restrict__ ei,
                           const float* __restrict__ als, const float* __restrict__ ald,
                           const unsigned* __restrict__ mbuf,
                           float* __restrict__ den) {
    int tid = blockIdx.x * blockDim.x + threadIdx.x;
    if (tid >= ETOT * HEADS_T) return;
    int e = tid / HEADS_T, h = tid % HEADS_T;
    int s, d; get_edge(ei, e, s, d);
    float ev = lrelu(als[s * HEADS_T + h] + ald[d * HEADS_T + h]);
    float m  = ord2f(mbuf[d * HEADS_T + h]);
    atomicAdd(&den[d * HEADS_T + h], expf(ev - m));
}

// aggregate: one thread per (edge, 4-channel group): b128 gather + 4 f32 atomics
template<int HEADS_T, int CPH>
__global__ void edge_agg4(const long long* __restrict__ ei,
                          const float* __restrict__ hsrc,
                          const float* __restrict__ als, const float* __restrict__ ald,
                          const unsigned* __restrict__ mbuf,
                          const float* __restrict__ den,
                          float* __restrict__ out) {
    constexpr int CH  = HEADS_T * CPH;
    constexpr int CH4 = CH / 4;
    long long tid = (long long)blockIdx.x * blockDim.x + threadIdx.x;
    if (tid >= (long long)ETOT * CH4) return;
    int e  = (int)(tid / CH4);
    int q  = (int)(tid % CH4);
    int c0 = q * 4;
    int h  = c0 / CPH;                    // CPH is a multiple of 4 -> group in one head
    int s, d; get_edge(ei, e, s, d);
    float ev    = lrelu(als[s * HEADS_T + h] + ald[d * HEADS_T + h]);
    float m     = ord2f(mbuf[d * HEADS_T + h]);
    float alpha = expf(ev - m) / (den[d * HEADS_T + h] + 1e-16f);
    const float4 hv = *(const float4*)(hsrc + (size_t)s * CH + c0);
    float* op = out + (size_t)d * CH + c0;
    atomicAdd(op + 0, hv.x * alpha);
    atomicAdd(op + 1, hv.y * alpha);
    atomicAdd(op + 2, hv.z * alpha);
    atomicAdd(op + 3, hv.w * alpha);
}

// ---------------- pointwise ----------------
__global__ void elu_bias_inplace(float* __restrict__ p, const float* __restrict__ b) {
    int i = blockIdx.x * blockDim.x + threadIdx.x;
    if (i >= NN * H1C) return;
    float v = p[i] + b[i & (H1C - 1)];
    p[i] = v > 0.f ? v : expm1f(v);
}

__global__ void add_bias2(float* __restrict__ p, const float* __restrict__ b) {
    int i = blockIdx.x * blockDim.x + threadIdx.x;
    if (i >= NN * OUTC) return;
    p[i] += b[i & (OUTC - 1)];
}

// ---------------- launch ----------------
extern "C" void kernel_launch(void* const* d_in, const int* in_sizes, int n_in,
                              void* d_out, int out_size, void* d_ws, size_t ws_size,
                              hipStream_t stream) {
    const float*     x   = (const float*)d_in[0];
    const long long* ei  = (const long long*)d_in[1];
    const float*     W1  = (const float*)d_in[2];
    const float*     a1s = (const float*)d_in[3];
    const float*     a1d = (const float*)d_in[4];
    const float*     b1  = (const float*)d_in[5];
    const float*     W2  = (const float*)d_in[6];
    const float*     a2s = (const float*)d_in[7];
    const float*     a2d = (const float*)d_in[8];
    const float*     b2  = (const float*)d_in[9];
    float*           out = (float*)d_out;

    // workspace layout (bytes)
    char* ws = (char*)d_ws;
    float*    h1   = (float*)(ws);                                // [NN,256]
    float*    out1 = (float*)(ws + 51200000);                     // [NN,256] agg, then ELU in place
    float*    h2   = (float*)(ws + 102400000);                    // [NN,64]
    float*    als1 = (float*)(ws + 115200000);                    // [NN,8]
    float*    ald1 = (float*)(ws + 116800000);
    unsigned* m1   = (unsigned*)(ws + 118400000);
    float*    den1 = (float*)(ws + 120000000);
    float*    als2 = (float*)(ws + 121600000);                    // [NN]
    float*    ald2 = (float*)(ws + 121800000);
    unsigned* m2   = (unsigned*)(ws + 122000000);
    float*    den2 = (float*)(ws + 122200000);

    const int B = 256;
    #define GRID(n) dim3(((n) + B - 1) / B)

    // ---- layer 1 ----
    {
        int mtiles = NN / 16;               // 3125
        int blocks = (mtiles + 3) / 4;      // 4 waves / 128-thread block
        gemm1_wmma<<<blocks, 128, 0, stream>>>(x, W1, h1, NN);
    }
    attn_logits1<<<GRID(NN * HEADS), B, 0, stream>>>(h1, a1s, a1d, als1, ald1);

    fill_u32<<<GRID(NN * HEADS), B, 0, stream>>>(m1, ORD_NEG_INF, NN * HEADS);
    fill_f32<<<GRID(NN * HEADS), B, 0, stream>>>(den1, 0.f, NN * HEADS);
    fill_f32<<<GRID(NN * H1C), B, 0, stream>>>(out1, 0.f, NN * H1C);

    edge_max<HEADS>  <<<GRID(ETOT * HEADS), B, 0, stream>>>(ei, als1, ald1, m1);
    edge_denom<HEADS><<<GRID(ETOT * HEADS), B, 0, stream>>>(ei, als1, ald1, m1, den1);
    {
        long long tot = (long long)ETOT * (H1C / 4);  // 54.4M threads
        int blocks = (int)((tot + B - 1) / B);
        edge_agg4<HEADS, HIDC><<<blocks, B, 0, stream>>>(ei, h1, als1, ald1, m1, den1, out1);
    }
    elu_bias_inplace<<<GRID(NN * H1C), B, 0, stream>>>(out1, b1);

    // ---- layer 2 ----
    {
        int mtiles = NN / 16;
        int blocks = (mtiles + 3) / 4;
        gemm2_wmma<<<blocks, 128, 0, stream>>>(out1, W2, h2, NN);
    }
    attn_logits2<<<GRID(NN), B, 0, stream>>>(h2, a2s, a2d, als2, ald2);

    fill_u32<<<GRID(NN), B, 0, stream>>>(m2, ORD_NEG_INF, NN);
    fill_f32<<<GRID(NN), B, 0, stream>>>(den2, 0.f, NN);
    fill_f32<<<GRID(NN * OUTC), B, 0, stream>>>(out, 0.f, NN * OUTC);

    edge_max<1>  <<<GRID(ETOT), B, 0, stream>>>(ei, als2, ald2, m2);
    edge_denom<1><<<GRID(ETOT), B, 0, stream>>>(ei, als2, ald2, m2, den2);
    {
        long long tot = (long long)ETOT * (OUTC / 4); // 13.6M threads
        int blocks = (int)((tot + B - 1) / B);
        edge_agg4<1, OUTC><<<blocks, B, 0, stream>>>(ei, h2, als2, ald2, m2, den2, out);
    }
    add_bias2<<<GRID(NN * OUTC), B, 0, stream>>>(out, b2);
    #undef GRID
}